// STELLA_154618823373
// MI455X (gfx1250) — compile-verified
//
#include <hip/hip_runtime.h>
#include <math.h>

// ---------------------------------------------------------------------------
// MI455X (gfx1250) implementation.
//
// N=8, H=12, DH=64, LV=1568, LA=512, NCV=392, NCA=256.
// ~40 MFLOP / ~100-150 MB gathered traffic -> HBM/gather bound (~5-10us at
// 23.3 TB/s).  f32 inputs -> matrix work uses the f32 WMMA path:
// V_WMMA_F32_16X16X4_F32 (D[16x16] = A[16x4] x B[4x16] + C).
//
// WMMA lane layouts (wave32):
//   A 16x4 f32 (documented): lane<16 = row M=lane, (K0,K1) in vgpr0/1;
//                            lane>=16 = row M=lane-16, (K2,K3).
//   B 4x16 f32 (assumed symmetric): lane<16 = col N=lane, rows (K0,K1);
//                                   lane>=16 = col N=lane-16, rows (K2,K3).
//   C/D 16x16 f32 (documented): lane<16 vgpr r = D[r][lane];
//                               lane>=16 vgpr r = D[8+r][lane-16].
// The vector operand is replicated across its free dimension so the 16x16
// result is replicated and trivially extracted.
//
// All gathers are branch-free: out-of-range tile lanes clamp their index to a
// valid row (their D rows are never stored), so loads stay unconditional
// v2f (global_load_b64) and LLVM can hoist them ahead of the WMMA chain.
// ---------------------------------------------------------------------------

typedef __attribute__((ext_vector_type(2))) float v2f;
typedef __attribute__((ext_vector_type(4))) float v4f;
typedef __attribute__((ext_vector_type(8))) float v8f;

#define N_    8
#define H_    12
#define DH_   64
#define LV_   1568
#define LA_   512
#define NCV_  392
#define NCA_  256
#define SCALE_ 0.125f        // DH^-0.5 = 1/8

// Output row layout (per n), stride 5696:
#define OS_     5696
#define OFF_PV  0            // prob_v_full   [0,1568)
#define OFF_PA  1568         // prob_a_full   [1568,2080)
#define OFF_CV  2080         // pos_v_cls     [2080,2848)
#define OFF_CA  2848         // pos_a_cls     [2848,3616)
#define OFF_RV  3616         // prune_n_av    [3616,5184)
#define OFF_RA  5184         // prune_n_va    [5184,5696)

// ---------------------------------------------------------------------------
// K1: gather per-token weights through the top-NC indices and reduce sums.
// ---------------------------------------------------------------------------
__global__ void k_gather_weights(const float* __restrict__ n_av,
                                 const float* __restrict__ n_va,
                                 const int*   __restrict__ ids_v,
                                 const int*   __restrict__ ids_a,
                                 float* __restrict__ wv, float* __restrict__ wa,
                                 float* __restrict__ Sav, float* __restrict__ Sva)
{
    __shared__ float red[256];
    const int n = blockIdx.x, t = threadIdx.x;

    float s = 0.f;
    for (int j = t; j < NCV_; j += blockDim.x) {
        float w = n_av[n * LV_ + ids_v[n * LV_ + j]];
        wv[n * NCV_ + j] = w;
        s += w;
    }
    red[t] = s; __syncthreads();
    for (int o = 128; o > 0; o >>= 1) { if (t < o) red[t] += red[t + o]; __syncthreads(); }
    if (t == 0) Sav[n] = red[0];
    __syncthreads();

    s = 0.f;
    for (int j = t; j < NCA_; j += blockDim.x) {
        float w = n_va[n * LA_ + ids_a[n * LA_ + j]];
        wa[n * NCA_ + j] = w;
        s += w;
    }
    red[t] = s; __syncthreads();
    for (int o = 128; o > 0; o >>= 1) { if (t < o) red[t] += red[t + o]; __syncthreads(); }
    if (t == 0) Sva[n] = red[0];
}

// ---------------------------------------------------------------------------
// K2: pos_*_cls[n,h,d] = sum_j Q[n,h,idx[j],d] * w[j] / S[n].
// One block per (n,h), thread d in [0,64): coalesced 256B gathered rows.
// ---------------------------------------------------------------------------
__global__ void k_cls(const float* __restrict__ Q, const int* __restrict__ ids,
                      const float* __restrict__ w, const float* __restrict__ S,
                      float* __restrict__ out, int L, int NC, int out_off)
{
    const int nh = blockIdx.x;
    const int n = nh / H_, h = nh % H_;
    const int d = threadIdx.x;
    const float* base = Q + (size_t)nh * L * DH_;
    float acc = 0.f;
    for (int j = 0; j < NC; ++j) {
        int r = ids[n * L + j];                    // uniform -> scalar load
        acc += base[(size_t)r * DH_ + d] * w[n * NC + j];
    }
    out[(size_t)n * OS_ + out_off + h * DH_ + d] = acc / S[n];
}

// ---------------------------------------------------------------------------
// K3 (WMMA): spu[n,h,l] = SCALE * sum_d K[n,h,ids[l],d] * cls[n,h,d].
// One wave per 16-row tile; A = gathered K rows, B = cls replicated over N.
// 16 accumulating V_WMMA_F32_16X16X4_F32 ops cover DH=64; all loads are
// unconditional 8-byte v2f so they can be hoisted/batched ahead of the
// accumulate chain.
// ---------------------------------------------------------------------------
__global__ void k_spu_wmma(const float* __restrict__ Kmat,
                           const float* __restrict__ cls,
                           const int*   __restrict__ ids,
                           float* __restrict__ spu, int L, int NC, int tiles)
{
    const int id   = blockIdx.x;
    const int tile = id % tiles;
    const int nh   = id / tiles;
    const int n    = nh / H_;
    const int lane = threadIdx.x;
    const bool hi  = lane >= 16;
    const int  m   = lane & 15;
    const int  l   = tile * 16 + m;

    // Clamp instead of masking: out-of-range lanes read a valid row whose
    // results are never stored.  Keeps loads branch-free.
    const int lc  = (l < NC) ? l : (NC - 1);
    const int row = ids[n * L + lc];
    const float* krow = Kmat + ((size_t)nh * L + row) * DH_;   // 256B-aligned row
    const float* c    = cls + (size_t)nh * DH_;
    const int sel = hi ? 2 : 0;

    v8f acc = {};
#pragma unroll
    for (int d0 = 0; d0 < DH_; d0 += 4) {
        const int off = d0 + sel;                               // even -> 8B aligned
        v2f aa = *reinterpret_cast<const v2f*>(krow + off);     // A row M = l
        v2f bb = *reinterpret_cast<const v2f*>(c + off);        // B replicated over N
        acc = __builtin_amdgcn_wmma_f32_16x16x4_f32(
                  false, aa, false, bb, (short)0, acc, false, false);
    }

    // D replicated across columns: lane0 holds M=0..7, lane16 holds M=8..15.
    // NC is a multiple of 8, so each 8-row group is all-valid or all-invalid.
    if (lane == 0 || lane == 16) {
        const int base = tile * 16 + (hi ? 8 : 0);
        if (base + 8 <= NC) {
            float* o = spu + (size_t)nh * NC + base;            // 32B aligned
            v4f lo = { acc[0] * SCALE_, acc[1] * SCALE_, acc[2] * SCALE_, acc[3] * SCALE_ };
            v4f hi4 = { acc[4] * SCALE_, acc[5] * SCALE_, acc[6] * SCALE_, acc[7] * SCALE_ };
            *reinterpret_cast<v4f*>(o)     = lo;
            *reinterpret_cast<v4f*>(o + 4) = hi4;
        }
    }
}

// ---------------------------------------------------------------------------
// K4 (WMMA): pos[n,h,l] = (1/S[n]) * sum_k Mat[n,h,row_ids[k],col_ids[l]] * w[k].
// One wave per 16-column tile; A = weight vector replicated over M rows,
// B = doubly-gathered 4x16 cross-attention tile.  NCrow is a multiple of 4.
// Prefetch of the next K-chunk's gathered row is branch-free (rids[k0+4] and
// the resulting address are provably inside the (n,h) slice).
// ---------------------------------------------------------------------------
__global__ void k_pos_wmma(const float* __restrict__ Mat,
                           const int*   __restrict__ row_ids,
                           const int*   __restrict__ col_ids,
                           const float* __restrict__ w,
                           const float* __restrict__ S,
                           float* __restrict__ pos,
                           int Lrow, int Lcol, int NCrow, int NCcol, int tiles)
{
    const int id   = blockIdx.x;
    const int tile = id % tiles;
    const int nh   = id / tiles;
    const int n    = nh / H_;
    const int lane = threadIdx.x;
    const bool hi  = lane >= 16;
    const int  m   = lane & 15;
    const int  l   = tile * 16 + m;
    const bool validc = l < NCcol;

    const int lcc  = validc ? l : (NCcol - 1);
    const int cidx = col_ids[n * Lcol + lcc];
    const float* Mbase = Mat + (size_t)nh * Lrow * Lcol;
    const int*   rids  = row_ids + n * Lrow;
    const float* wn    = w + n * NCrow;                         // 8B aligned
    const int sel = hi ? 2 : 0;

    v8f acc = {};
#pragma unroll 4
    for (int a0 = 0; a0 < NCrow; a0 += 4) {
        const int k0 = a0 + sel;                                // even -> 8B aligned
        v2f aa = *reinterpret_cast<const v2f*>(wn + k0);        // A replicated over M
        const int r0 = rids[k0], r1 = rids[k0 + 1];
        v2f bb;
        bb.x = Mbase[(size_t)r0 * Lcol + cidx];                 // gathered 4x16 B tile
        bb.y = Mbase[(size_t)r1 * Lcol + cidx];
        // Branch-free lookahead: k0+4 <= NCrow+2 < Lrow, value < Lrow.
        __builtin_prefetch(&Mbase[(size_t)rids[k0 + 4] * Lcol + cidx], 0, 0);
        acc = __builtin_amdgcn_wmma_f32_16x16x4_f32(
                  false, aa, false, bb, (short)0, acc, false, false);
    }

    // D replicated across rows: every lane's acc[0] is its column's answer.
    if (!hi && validc)
        pos[(size_t)nh * NCcol + l] = acc[0] / S[n];
}

// ---------------------------------------------------------------------------
// K5a: default-fill output (prob regions zero, prune regions = n_attn).
// ---------------------------------------------------------------------------
__global__ void k_init_out(const float* __restrict__ n_av,
                           const float* __restrict__ n_va,
                           float* __restrict__ out)
{
    const int n = blockIdx.x, t = threadIdx.x;
    float* o = out + (size_t)n * OS_;
    for (int i = t; i < LV_; i += blockDim.x) {
        o[OFF_PV + i] = 0.f;
        o[OFF_RV + i] = n_av[n * LV_ + i];
    }
    for (int i = t; i < LA_; i += blockDim.x) {
        o[OFF_PA + i] = 0.f;
        o[OFF_RA + i] = n_va[n * LA_ + i];
    }
}

// ---------------------------------------------------------------------------
// K5b: prob = mean_h sigmoid(spu - pos); scatter through the permutation and
// apply conditional prune at the scattered positions.
// ---------------------------------------------------------------------------
__global__ void k_prob_scatter(const float* __restrict__ spu,
                               const float* __restrict__ posm,
                               const int*   __restrict__ ids,
                               const float* __restrict__ u,
                               const float* __restrict__ n_attn,
                               float* __restrict__ out,
                               int L, int NC, int off_p, int off_r)
{
    const int n = blockIdx.x;
    float* o = out + (size_t)n * OS_;
    for (int j = threadIdx.x; j < NC; j += blockDim.x) {
        float p = 0.f;
#pragma unroll
        for (int h = 0; h < H_; ++h) {
            const float s = spu[((size_t)n * H_ + h) * NC + j];
            const float q = posm[((size_t)n * H_ + h) * NC + j];
            p += 1.f / (1.f + __expf(q - s));   // softmax([s,q])[0]
        }
        p *= (1.f / H_);
        const int tpos = ids[n * L + j];
        o[off_p + tpos] = p;
        o[off_r + tpos] = (u[n * L + tpos] < p) ? 0.f : n_attn[n * L + tpos];
    }
}

// ---------------------------------------------------------------------------
extern "C" void kernel_launch(void* const* d_in, const int* in_sizes, int n_in,
                              void* d_out, int out_size, void* d_ws, size_t ws_size,
                              hipStream_t stream)
{
    const float* pos_v_q   = (const float*)d_in[0];
    const float* pos_v_k   = (const float*)d_in[1];
    const float* pos_a_q   = (const float*)d_in[2];
    const float* pos_a_k   = (const float*)d_in[3];
    const float* cross_av  = (const float*)d_in[4];   // (N,H,LA,LV)
    const float* cross_va  = (const float*)d_in[5];   // (N,H,LV,LA)
    const float* n_av      = (const float*)d_in[6];
    const float* n_va      = (const float*)d_in[7];
    const float* spu_a_cls = (const float*)d_in[8];
    const float* spu_v_cls = (const float*)d_in[9];
    const float* u_v       = (const float*)d_in[10];
    const float* u_a       = (const float*)d_in[11];
    const int*   ids_v     = (const int*)d_in[12];    // att_av_ids (N,LV)
    const int*   ids_a     = (const int*)d_in[13];    // att_va_ids (N,LA)
    float* out = (float*)d_out;

    // Workspace carve-up (~520 KB of f32).
    float* ws   = (float*)d_ws;
    float* wv   = ws;  ws += N_ * NCV_;
    float* wa   = ws;  ws += N_ * NCA_;
    float* Sav  = ws;  ws += N_;
    float* Sva  = ws;  ws += N_;
    float* vspu = ws;  ws += N_ * H_ * NCV_;
    float* vpos = ws;  ws += N_ * H_ * NCV_;
    float* aspu = ws;  ws += N_ * H_ * NCA_;
    float* apos = ws;  ws += N_ * H_ * NCA_;

    const int TV = (NCV_ + 15) / 16;   // 25 column tiles
    const int TA = (NCA_ + 15) / 16;   // 16 column tiles

    k_gather_weights<<<N_, 256, 0, stream>>>(n_av, n_va, ids_v, ids_a, wv, wa, Sav, Sva);

    k_cls<<<N_ * H_, DH_, 0, stream>>>(pos_v_q, ids_v, wv, Sav, out, LV_, NCV_, OFF_CV);
    k_cls<<<N_ * H_, DH_, 0, stream>>>(pos_a_q, ids_a, wa, Sva, out, LA_, NCA_, OFF_CA);

    k_spu_wmma<<<N_ * H_ * TV, 32, 0, stream>>>(pos_v_k, spu_a_cls, ids_v, vspu, LV_, NCV_, TV);
    k_spu_wmma<<<N_ * H_ * TA, 32, 0, stream>>>(pos_a_k, spu_v_cls, ids_a, aspu, LA_, NCA_, TA);

    // v_pos: rows gathered by ids_a (K=NCA), cols by ids_v, weights d_n_va.
    k_pos_wmma<<<N_ * H_ * TV, 32, 0, stream>>>(cross_av, ids_a, ids_v, wa, Sva, vpos,
                                                LA_, LV_, NCA_, NCV_, TV);
    // a_pos: rows gathered by ids_v (K=NCV), cols by ids_a, weights d_n_av.
    k_pos_wmma<<<N_ * H_ * TA, 32, 0, stream>>>(cross_va, ids_v, ids_a, wv, Sav, apos,
                                                LV_, LA_, NCV_, NCA_, TA);

    k_init_out<<<N_, 256, 0, stream>>>(n_av, n_va, out);
    k_prob_scatter<<<N_, 256, 0, stream>>>(vspu, vpos, ids_v, u_v, n_av, out,
                                           LV_, NCV_, OFF_PV, OFF_RV);
    k_prob_scatter<<<N_, 256, 0, stream>>>(aspu, apos, ids_a, u_a, n_va, out,
                                           LA_, NCA_, OFF_PA, OFF_RA);
}